// GraphAttentionLayer_39762807226535
// MI455X (gfx1250) — compile-verified
//
#include <hip/hip_runtime.h>
#include <math.h>

#define BB 4
#define NN 4096
#define CC 64
#define HH 4
#define DD 32
#define KK 16
#define LRELU 0.2f
#define BIGF 1.0e9f
#define SENT 3.0e38f

typedef __attribute__((ext_vector_type(2))) float v2f;
typedef __attribute__((ext_vector_type(8))) float v8f;

__device__ __forceinline__ bool dless(float d1, int i1, float d2, int i2) {
  return (d1 < d2) || (d1 == d2 && i1 < i2);
}

// ---------------------------------------------------------------------------
// Kernel 1: h = x@W, sq = sum(x*x), e_src = h . a[:D], e_dst = h . a[D:]
// grid = B*N blocks, 128 threads (one output column each)
// ---------------------------------------------------------------------------
__global__ void __launch_bounds__(128)
prep_kernel(const float* __restrict__ x, const float* __restrict__ W,
            const float* __restrict__ a, float* __restrict__ h,
            float* __restrict__ sq, float* __restrict__ esrc,
            float* __restrict__ edst)
{
  const int n = blockIdx.x;   // global node over B*N
  const int t = threadIdx.x;
  __shared__ float xs[CC];
  __shared__ float hs[HH * DD];

  if (t < CC) xs[t] = x[(size_t)n * CC + t];
  __syncthreads();

  float acc = 0.0f;
#pragma unroll 8
  for (int c = 0; c < CC; ++c) acc += xs[c] * W[c * (HH * DD) + t];
  h[(size_t)n * (HH * DD) + t] = acc;
  hs[t] = acc;
  __syncthreads();

  if (t < HH) {
    float s = 0.0f;
    for (int d = 0; d < DD; ++d) s += hs[t * DD + d] * a[d];
    esrc[(size_t)n * HH + t] = s;
  } else if (t < 2 * HH) {
    const int hh = t - HH;
    float s = 0.0f;
    for (int d = 0; d < DD; ++d) s += hs[hh * DD + d] * a[DD + d];
    edst[(size_t)n * HH + hh] = s;
  } else if (t == 2 * HH) {
    float s = 0.0f;
    for (int c = 0; c < CC; ++c) s += xs[c] * xs[c];
    sq[n] = s;
  }
}

// ---------------------------------------------------------------------------
// Kernel 2: fused pairwise-distance (WMMA f32 16x16x4) + top-16 KNN.
// Ranking is done on clamped squared distance (sqrt is monotonic -> same
// neighbor set as the reference, no sqrt in the hot loop).
// grid = B*(N/16) workgroups of 128 threads (4 waves).
// ---------------------------------------------------------------------------
__global__ void __launch_bounds__(128)
knn_kernel(const float* __restrict__ x, const unsigned char* __restrict__ mask,
           const float* __restrict__ sq, int* __restrict__ knn)
{
  __shared__ float sds[16 * 512];     // 32 KB distance chunk
  __shared__ float sqQ[16];
  __shared__ int   mQ[16];
  __shared__ float md[16][8][16];
  __shared__ int   mi[16][8][16];

  const int t    = threadIdx.x;
  const int lane = t & 31;
  const int w    = t >> 5;            // wave id 0..3 (wave32)
  const int b    = blockIdx.x >> 8;
  const int q0   = (blockIdx.x & 255) << 4;

  const int qrow = lane & 15;         // M (or N) index within tile
  const int kh   = lane >> 4;         // lane-half selects K pair / row-half

  const float* xb  = x  + (size_t)b * NN * CC;
  const float* sqb = sq + (size_t)b * NN;
  const unsigned char* mb = mask + (size_t)b * NN;

  if (t < 16) {
    sqQ[t] = sqb[q0 + t];
    mQ[t]  = mb[q0 + t] ? 1 : 0;
  }

  // A operand (query tile) resident in registers: 16 K-steps of 4.
  // f32 A 16x4 layout: VGPR0 = {K=k0 | K=k0+2}, VGPR1 = {K=k0+1 | K=k0+3}
  v2f A[16];
  const float* xq = xb + (size_t)(q0 + qrow) * CC;
#pragma unroll
  for (int kk = 0; kk < 16; ++kk) {
    A[kk].x = xq[4 * kk + 2 * kh];
    A[kk].y = xq[4 * kk + 2 * kh + 1];
  }

  float ld[16]; int li[16];
#pragma unroll
  for (int p = 0; p < 16; ++p) { ld[p] = SENT; li[p] = 0x7fffffff; }

  const int qr = t >> 3;              // query this thread selects for
  const int s  = t & 7;               // slot 0..7 within query

  for (int chunk = 0; chunk < NN / 512; ++chunk) {
    const int cb = chunk * 512;
    __syncthreads();                  // sds free to overwrite

#pragma unroll 1
    for (int tile = 0; tile < 8; ++tile) {
      const int c0 = cb + w * 128 + tile * 16;
      const float* xc = xb + (size_t)(c0 + qrow) * CC;

      // prefetch next tile's candidate rows (global_prefetch_b8)
      if (tile < 7) {
        const float* xn = xc + 16 * CC;
        __builtin_prefetch(xn, 0, 3);
        __builtin_prefetch((const char*)xn + 128, 0, 3);
      }

      // two independent accumulator chains for WMMA ILP
      v8f acc0 = {0.f, 0.f, 0.f, 0.f, 0.f, 0.f, 0.f, 0.f};
      v8f acc1 = {0.f, 0.f, 0.f, 0.f, 0.f, 0.f, 0.f, 0.f};
#pragma unroll
      for (int kk = 0; kk < 16; kk += 2) {
        v2f B0, B1;
        B0.x = xc[4 * kk + 2 * kh];
        B0.y = xc[4 * kk + 2 * kh + 1];
        B1.x = xc[4 * (kk + 1) + 2 * kh];
        B1.y = xc[4 * (kk + 1) + 2 * kh + 1];
        acc0 = __builtin_amdgcn_wmma_f32_16x16x4_f32(
            false, A[kk], false, B0, (short)0, acc0, false, false);
        acc1 = __builtin_amdgcn_wmma_f32_16x16x4_f32(
            false, A[kk + 1], false, B1, (short)0, acc1, false, false);
      }
      const float sqC = sqb[c0 + qrow];
      const int   mC  = mb[c0 + qrow] ? 1 : 0;
      const int colLocal = w * 128 + tile * 16 + qrow;
#pragma unroll
      for (int r = 0; r < 8; ++r) {
        const int row = r + kh * 8;   // D layout: lanes16-31 hold M=r+8
        const float dot = acc0[r] + acc1[r];
        float dd = fmaxf(sqQ[row] + sqC - 2.0f * dot, 0.0f); // rank on d2
        if (!(mQ[row] && mC)) dd = BIGF;
        if (c0 + qrow == q0 + row) dd = SENT;  // exclude self
        sds[row * 512 + colLocal] = dd;
      }
    }
    __syncthreads();

    // per-thread top-16 maintenance over 64 candidates of this chunk
#pragma unroll 1
    for (int i = 0; i < 64; ++i) {
      const int j = s + 8 * i;
      const float dd = sds[qr * 512 + j];
      const int cid = cb + j;
      if (dless(dd, cid, ld[15], li[15])) {
        ld[15] = dd; li[15] = cid;
#pragma unroll
        for (int p = 15; p > 0; --p) {
          if (dless(ld[p], li[p], ld[p - 1], li[p - 1])) {
            float td = ld[p]; ld[p] = ld[p - 1]; ld[p - 1] = td;
            int   ti = li[p]; li[p] = li[p - 1]; li[p - 1] = ti;
          }
        }
      }
    }
  }

  __syncthreads();
#pragma unroll
  for (int p = 0; p < 16; ++p) { md[qr][s][p] = ld[p]; mi[qr][s][p] = li[p]; }
  __syncthreads();

  // 8-way merge of sorted lists: threads 0..15 (one per query)
  if (t < 16) {
    int head[8] = {0, 0, 0, 0, 0, 0, 0, 0};
    int* kout = knn + ((size_t)b * NN + q0 + t) * KK;
    for (int pick = 0; pick < KK; ++pick) {
      float bd = INFINITY; int bi = 0x7fffffff; int bs = 0;
      for (int s2 = 0; s2 < 8; ++s2) {
        const int hp = head[s2];
        if (hp < 16) {
          const float dd = md[t][s2][hp];
          const int   ii = mi[t][s2][hp];
          if (dless(dd, ii, bd, bi)) { bd = dd; bi = ii; bs = s2; }
        }
      }
      head[bs]++;
      kout[pick] = bi;
    }
  }
}

// ---------------------------------------------------------------------------
// Kernel 3: gather neighbors, leaky-relu, softmax over K, weighted sum.
// grid = B*N blocks of 128 threads (thread t -> head t/32, dim t%32).
// ---------------------------------------------------------------------------
__global__ void __launch_bounds__(128)
attn_kernel(const float* __restrict__ h, const float* __restrict__ esrc,
            const float* __restrict__ edst, const int* __restrict__ knn,
            float* __restrict__ out)
{
  const int n = blockIdx.x;           // global node over B*N
  const int b = n >> 12;              // n / NN
  const int t = threadIdx.x;
  const int head = t >> 5;

  __shared__ int nb[KK];
  __shared__ float al[HH][KK];

  if (t < KK) nb[t] = knn[(size_t)n * KK + t];
  __syncthreads();

  if (t < HH) {
    const float es = esrc[(size_t)n * HH + t];
    float e[KK];
    float m = -INFINITY;
#pragma unroll
    for (int k = 0; k < KK; ++k) {
      float v = es + edst[((size_t)b * NN + nb[k]) * HH + t];
      v = (v >= 0.0f) ? v : LRELU * v;
      e[k] = v;
      m = fmaxf(m, v);
    }
    float sum = 0.0f;
#pragma unroll
    for (int k = 0; k < KK; ++k) { e[k] = expf(e[k] - m); sum += e[k]; }
    const float inv = 1.0f / sum;
#pragma unroll
    for (int k = 0; k < KK; ++k) al[t][k] = e[k] * inv;
  }
  __syncthreads();

  float acc = 0.0f;
#pragma unroll
  for (int k = 0; k < KK; ++k)
    acc += al[head][k] * h[((size_t)b * NN + nb[k]) * (HH * DD) + t];
  out[(size_t)n * (HH * DD) + t] = acc;
}

// ---------------------------------------------------------------------------
extern "C" void kernel_launch(void* const* d_in, const int* in_sizes, int n_in,
                              void* d_out, int out_size, void* d_ws, size_t ws_size,
                              hipStream_t stream) {
  (void)in_sizes; (void)n_in; (void)out_size; (void)ws_size;
  const float*         x    = (const float*)d_in[0];
  const unsigned char* mask = (const unsigned char*)d_in[1];
  const float*         W    = (const float*)d_in[2];
  const float*         a    = (const float*)d_in[3];
  float* out = (float*)d_out;

  float* h    = (float*)d_ws;                          // B*N*H*D
  float* sq   = h    + (size_t)BB * NN * HH * DD;      // B*N
  float* esrc = sq   + (size_t)BB * NN;                // B*N*H
  float* edst = esrc + (size_t)BB * NN * HH;           // B*N*H
  int*   knn  = (int*)(edst + (size_t)BB * NN * HH);   // B*N*K

  prep_kernel<<<BB * NN, 128, 0, stream>>>(x, W, a, h, sq, esrc, edst);
  knn_kernel<<<BB * (NN / 16), 128, 0, stream>>>(x, mask, sq, knn);
  attn_kernel<<<BB * NN, 128, 0, stream>>>(h, esrc, edst, knn, out);
}